// ChamferLoss_60756607369675
// MI455X (gfx1250) — compile-verified
//
#include <hip/hip_runtime.h>

typedef float v2f __attribute__((ext_vector_type(2)));
typedef float v8f __attribute__((ext_vector_type(8)));

#define BATCH   4
#define NPTS    8192
#define MPTS    8192
#define TILES_PER_WAVE 4                 // 4 x 16 = 64 rows per wave
#define THREADS 256                      // 8 waves
#define ROWS_PER_BLOCK (64 * (THREADS / 32))            // 512
#define BLOCKS_PER_BATCH (NPTS / ROWS_PER_BLOCK)        // 16
#define BLOCKS_PER_DIR (BATCH * BLOCKS_PER_BATCH)       // 64
#define TOTAL_BLOCKS (2 * BLOCKS_PER_DIR)               // 128
#define FLT_BIG 3.402823466e38f

// Both chamfer directions in one launch: blocks [0,64) do xyz1->xyz2,
// blocks [64,128) do xyz2->xyz1.  Each wave owns 4 sixteen-row A tiles and
// sweeps M in 32-column steps (2 B fragments), issuing 8 WMMAs per step:
//   A row i  = (-2px, -2py, -2pz, 1)
//   B col j  = ( qx,  qy,  qz, ||q||^2 )
//   D[i][j]  = -2 p.q + ||q||^2 ;  add ||p||^2 per row after the min.
// The two D tiles per A tile are folded into the running min with
// fminf(fminf(rm,d0),d1) so the backend emits one v_min3_num_f32 per element.
__global__ __launch_bounds__(THREADS)
void chamfer_both_kernel(const float* __restrict__ xyz1,
                         const float* __restrict__ xyz2,
                         float* __restrict__ partial)
{
    const int dir = (blockIdx.x < BLOCKS_PER_DIR) ? 0 : 1;
    const float* P = dir ? xyz2 : xyz1;
    const float* Q = dir ? xyz1 : xyz2;
    const int blk = blockIdx.x - dir * BLOCKS_PER_DIR;
    const int b        = blk / BLOCKS_PER_BATCH;
    const int rowGroup = blk % BLOCKS_PER_BATCH;

    const int lane = threadIdx.x & 31;
    const int wave = threadIdx.x >> 5;
    const bool hi  = lane >= 16;
    const int  l16 = lane & 15;
    const int rowBase = rowGroup * ROWS_PER_BLOCK + wave * (16 * TILES_PER_WAVE);

    // ---- 4 A fragments (16x4 f32 each): lanes 0-15 carry K=0,1; lanes 16-31 K=2,3
    v2f afrag[TILES_PER_WAVE];
#pragma unroll
    for (int t = 0; t < TILES_PER_WAVE; ++t) {
        const float* pr = P + ((size_t)b * NPTS + rowBase + t * 16 + l16) * 3;
        const float px = pr[0], py = pr[1], pz = pr[2];
        afrag[t].x = hi ? (-2.0f * pz) : (-2.0f * px);   // K=2 : K=0
        afrag[t].y = hi ? 1.0f          : (-2.0f * py);  // K=3 : K=1
    }

    // running mins: rm[t][r] ~ D VGPR r of tile t (rows r / r+8 of that tile)
    float rm[TILES_PER_WAVE][8];
#pragma unroll
    for (int t = 0; t < TILES_PER_WAVE; ++t)
#pragma unroll
        for (int r = 0; r < 8; ++r) rm[t][r] = FLT_BIG;

    const float* Qb = Q + (size_t)b * MPTS * 3;
    for (int colBase = 0; colBase < MPTS; colBase += 32) {
        // ---- two B fragments (4x16 f32), amortized over 8 WMMAs
        const float* qc0 = Qb + (size_t)(colBase + l16) * 3;
        const float x0 = qc0[0], y0 = qc0[1], z0 = qc0[2];
        const float* qc1 = Qb + (size_t)(colBase + 16 + l16) * 3;
        const float x1 = qc1[0], y1 = qc1[1], z1 = qc1[2];
        const float n0 = x0 * x0 + y0 * y0 + z0 * z0;
        const float n1 = x1 * x1 + y1 * y1 + z1 * z1;
        v2f bf0, bf1;
        bf0.x = hi ? z0 : x0;   bf0.y = hi ? n0 : y0;
        bf1.x = hi ? z1 : x1;   bf1.y = hi ? n1 : y1;

#pragma unroll
        for (int t = 0; t < TILES_PER_WAVE; ++t) {
            v8f c = {};
            v8f d0 = __builtin_amdgcn_wmma_f32_16x16x4_f32(
                false, afrag[t], false, bf0, (short)0, c, false, false);
            v8f d1 = __builtin_amdgcn_wmma_f32_16x16x4_f32(
                false, afrag[t], false, bf1, (short)0, c, false, false);
#pragma unroll
            for (int r = 0; r < 8; ++r)
                rm[t][r] = fminf(fminf(rm[t][r], d0[r]), d1[r]);  // -> v_min3_num
        }
    }

    // ---- row-min: reduce across the 16 lanes of each half-wave (wave32)
#pragma unroll
    for (int m = 1; m < 16; m <<= 1)
#pragma unroll
        for (int t = 0; t < TILES_PER_WAVE; ++t)
#pragma unroll
            for (int r = 0; r < 8; ++r)
                rm[t][r] = fminf(rm[t][r], __shfl_xor(rm[t][r], m, 16));

    // lanes 0 / 16 hold mins for rows {t*16 + 0..7} / {t*16 + 8..15}.
    // Add each row's ||p||^2 and accumulate this wave-half's sum.
    __shared__ float sdata[2 * (THREADS / 32)];
    if (l16 == 0) {
        float s = 0.0f;
#pragma unroll
        for (int t = 0; t < TILES_PER_WAVE; ++t) {
            const int rowOff = rowBase + t * 16 + (hi ? 8 : 0);
#pragma unroll
            for (int r = 0; r < 8; ++r) {
                const float* pr = P + ((size_t)b * NPTS + rowOff + r) * 3;
                const float x = pr[0], y = pr[1], z = pr[2];
                s += rm[t][r] + (x * x + y * y + z * z);
            }
        }
        sdata[wave * 2 + (hi ? 1 : 0)] = s;
    }
    __syncthreads();
    if (threadIdx.x == 0) {
        float s = 0.0f;
#pragma unroll
        for (int i = 0; i < 2 * (THREADS / 32); ++i) s += sdata[i];
        partial[blockIdx.x] = s;   // every block writes -> no init pass needed
    }
}

// Deterministic final reduction: fixed-order tree over the 128 block partials.
__global__ __launch_bounds__(TOTAL_BLOCKS)
void chamfer_finalize_kernel(const float* __restrict__ ws, float* __restrict__ out,
                             float invBN, float invBM)
{
    __shared__ float sdata[TOTAL_BLOCKS];
    const int t = threadIdx.x;
    sdata[t] = ws[t] * ((t < BLOCKS_PER_DIR) ? invBN : invBM);
    __syncthreads();
    for (int off = TOTAL_BLOCKS / 2; off > 0; off >>= 1) {
        if (t < off) sdata[t] += sdata[t + off];
        __syncthreads();
    }
    if (t == 0) out[0] = sdata[0];
}

extern "C" void kernel_launch(void* const* d_in, const int* in_sizes, int n_in,
                              void* d_out, int out_size, void* d_ws, size_t ws_size,
                              hipStream_t stream)
{
    (void)in_sizes; (void)n_in; (void)out_size; (void)ws_size;
    const float* xyz1 = (const float*)d_in[0];   // (B, N, 3) f32
    const float* xyz2 = (const float*)d_in[1];   // (B, M, 3) f32
    float* ws  = (float*)d_ws;                   // [0..127] block partials
    float* out = (float*)d_out;

    chamfer_both_kernel<<<TOTAL_BLOCKS, THREADS, 0, stream>>>(xyz1, xyz2, ws);
    chamfer_finalize_kernel<<<1, TOTAL_BLOCKS, 0, stream>>>(
        ws, out, 1.0f / (float)(BATCH * NPTS), 1.0f / (float)(BATCH * MPTS));
}